// CentralityEncoding_38517266710630
// MI455X (gfx1250) — compile-verified
//
#include <hip/hip_runtime.h>
#include <hip/hip_bf16.h>

#define NUM_NODES 50000
#define NUM_EDGES 1600000
#define NODE_DIM 256
#define EDGE_DIM 128
#define MAX_DEG 512

typedef float v2f __attribute__((ext_vector_type(2)));
typedef float v4f __attribute__((ext_vector_type(4)));
typedef float v8f __attribute__((ext_vector_type(8)));

// ---------------------------------------------------------------------------
// Phase 1: zero the scratch accumulator + degree counters
// ---------------------------------------------------------------------------
__global__ void zero_ws_kernel(float* __restrict__ agg, int* __restrict__ cnts,
                               int n_agg, int n_cnt) {
    int i = blockIdx.x * blockDim.x + threadIdx.x;
    int stride = gridDim.x * blockDim.x;
    for (int j = i; j < n_agg; j += stride) agg[j] = 0.0f;
    for (int j = i; j < n_cnt; j += stride) cnts[j] = 0;
}

// ---------------------------------------------------------------------------
// Phase 2: scatter edge_attr rows into agg[src] (atomics land in L2: 25.6MB
// accumulator << 192MB L2), and count in/out degrees.
// One 32-lane wave per edge; each lane handles 4 contiguous dims (16B).
// edge_attr (819MB) is streamed once -> non-temporal loads so it does not
// evict the L2-resident accumulator.
// ---------------------------------------------------------------------------
__global__ __launch_bounds__(256)
void scatter_kernel(const v4f* __restrict__ edge_attr4,
                    const int* __restrict__ edge_index,
                    float* __restrict__ agg,
                    int* __restrict__ out_cnt,
                    int* __restrict__ in_cnt) {
    long long tid = (long long)blockIdx.x * blockDim.x + threadIdx.x;
    int e    = (int)(tid >> 5);
    int part = (int)(tid & 31);
    if (e >= NUM_EDGES) return;

    int src = edge_index[e];                 // row 0 of edge_index
    if (part == 0) atomicAdd(&out_cnt[src], 1);
    if (part == 1) atomicAdd(&in_cnt[edge_index[NUM_EDGES + e]], 1);  // row 1 = dst

    v4f v = __builtin_nontemporal_load(&edge_attr4[(long long)e * 32 + part]);
    float* dst = agg + (long long)src * EDGE_DIM + part * 4;
    atomicAdd(dst + 0, v.x);
    atomicAdd(dst + 1, v.y);
    atomicAdd(dst + 2, v.z);
    atomicAdd(dst + 3, v.w);
}

// ---------------------------------------------------------------------------
// Phase 3: WMMA fp32 GEMM  agg[50000x128] @ W_sum[128x256]  + fused epilogue.
// One wave per 16x16 output tile. 50000 rows = 3125 M-tiles exactly; 16
// N-tiles; 6250 blocks * 8 waves = 50000 tiles exactly -> EXEC all-ones.
//
// f32 WMMA 16x16x4 VGPR layouts (ISA 7.12.2):
//   A (16x4):  lanes 0-15 hold rows M=0..15 with VGPR0=K0,VGPR1=K1;
//              lanes 16-31 hold K2,K3.
//   B (4x16):  lanes 0-15 hold cols N=0..15 with VGPR0=K0,VGPR1=K1;
//              lanes 16-31 hold K2,K3.
//   C/D 16x16: VGPR r = row r (lanes 0-15) / row r+8 (lanes 16-31), N = lane%16.
// ---------------------------------------------------------------------------
__global__ __launch_bounds__(256)
void gemm_epilogue_kernel(const float* __restrict__ agg,
                          const float* __restrict__ W_sum,
                          const float* __restrict__ x,
                          const float* __restrict__ z_in,
                          const float* __restrict__ z_out,
                          const float* __restrict__ cent,
                          const float* __restrict__ vor,
                          const float* __restrict__ W_cent,
                          const float* __restrict__ b_cent,
                          const float* __restrict__ W_vor,
                          const float* __restrict__ b_vor,
                          const float* __restrict__ b_sum,
                          const int* __restrict__ in_cnt,
                          const int* __restrict__ out_cnt,
                          float* __restrict__ out) {
    int lane = threadIdx.x & 31;
    int wave = threadIdx.x >> 5;
    int tile = blockIdx.x * 8 + wave;       // [0, 50000)
    int nTile = tile & 15;                  // 16 N tiles
    int mTile = tile >> 4;                  // 3125 M tiles
    int mBase = mTile * 16;
    int nBase = nTile * 16;
    int half  = lane >> 4;                  // 0: K{0,1}, 1: K{2,3}
    int l16   = lane & 15;

    // A: this lane streams row (mBase+l16), K pairs at half*2
    const float* arow = agg + (long long)(mBase + l16) * EDGE_DIM + half * 2;
    // B: this lane streams column (nBase+l16), K pairs at half*2 (row-major W_sum)
    const float* bcol = W_sum + (long long)(half * 2) * NODE_DIM + nBase + l16;

    v8f c = {};
#pragma unroll
    for (int kk = 0; kk < EDGE_DIM / 4; ++kk) {
        v2f a = *(const v2f*)(arow + kk * 4);           // 8B-aligned pair load
        v2f b;
        b.x = bcol[kk * 4 * NODE_DIM];
        b.y = bcol[kk * 4 * NODE_DIM + NODE_DIM];
        // 8 args: (neg_a, A, neg_b, B, c_mod, C, reuse_a, reuse_b)
        c = __builtin_amdgcn_wmma_f32_16x16x4_f32(
                false, a, false, b, (short)0, c, false, false);
    }

    // Fused epilogue: out = x + z_in[clamp(in)] + z_out[clamp(out)]
    //                     + sigmoid(cent*Wc+bc) + sigmoid(vor*Wv+bv)
    //                     + agg@W_sum + raw_out_count*b_sum
    int   col = nBase + l16;
    float wc = W_cent[col], bc = b_cent[col];
    float wv = W_vor[col],  bv = b_vor[col];
    float bs = b_sum[col];

#pragma unroll
    for (int r = 0; r < 8; ++r) {
        int row = mBase + r + half * 8;
        int ic  = in_cnt[row];
        int oc  = out_cnt[row];
        int icl = ic < (MAX_DEG - 1) ? ic : (MAX_DEG - 1);
        int ocl = oc < (MAX_DEG - 1) ? oc : (MAX_DEG - 1);
        float s1 = 1.0f / (1.0f + __expf(-(cent[row] * wc + bc)));
        float s2 = 1.0f / (1.0f + __expf(-(vor[row]  * wv + bv)));
        float xv = __builtin_nontemporal_load(&x[row * NODE_DIM + col]);
        float val = xv
                  + z_in[icl * NODE_DIM + col]
                  + z_out[ocl * NODE_DIM + col]
                  + s1 + s2
                  + c[r]
                  + (float)oc * bs;
        __builtin_nontemporal_store(val, &out[row * NODE_DIM + col]);
    }
}

// ---------------------------------------------------------------------------
extern "C" void kernel_launch(void* const* d_in, const int* in_sizes, int n_in,
                              void* d_out, int out_size, void* d_ws, size_t ws_size,
                              hipStream_t stream) {
    const float* x          = (const float*)d_in[0];
    const int*   edge_index = (const int*)  d_in[1];
    const float* edge_attr  = (const float*)d_in[2];
    const float* vor        = (const float*)d_in[3];
    const float* cent       = (const float*)d_in[4];
    const float* z_in       = (const float*)d_in[5];
    const float* z_out      = (const float*)d_in[6];
    const float* W_cent     = (const float*)d_in[7];
    const float* b_cent     = (const float*)d_in[8];
    const float* W_vor      = (const float*)d_in[9];
    const float* b_vor      = (const float*)d_in[10];
    const float* W_sum      = (const float*)d_in[11];
    const float* b_sum      = (const float*)d_in[12];
    float* out = (float*)d_out;

    // Workspace layout: agg[50000*128] f32 | in_cnt[50000] i32 | out_cnt[50000] i32
    float* agg    = (float*)d_ws;
    int*   in_cnt = (int*)((char*)d_ws + (size_t)NUM_NODES * EDGE_DIM * sizeof(float));
    int*   out_cnt = in_cnt + NUM_NODES;

    // Phase 1: zero scratch (agg + both counters; counters are contiguous)
    zero_ws_kernel<<<2048, 256, 0, stream>>>(agg, in_cnt,
                                             NUM_NODES * EDGE_DIM, 2 * NUM_NODES);

    // Phase 2: scatter + degree counts. 1.6M edges * 32 lanes = 51.2M threads.
    long long scat_threads = (long long)NUM_EDGES * 32;
    int scat_blocks = (int)((scat_threads + 255) / 256);   // = 200000 exactly
    scatter_kernel<<<scat_blocks, 256, 0, stream>>>(
        (const v4f*)edge_attr, edge_index, agg, out_cnt, in_cnt);

    // Phase 3: WMMA GEMM + epilogue. 50000 tiles / 8 waves = 6250 blocks exactly.
    gemm_epilogue_kernel<<<6250, 256, 0, stream>>>(
        agg, W_sum, x, z_in, z_out, cent, vor,
        W_cent, b_cent, W_vor, b_vor, b_sum,
        in_cnt, out_cnt, out);
}